// NIMSCrossEntropyLoss_57294863728909
// MI455X (gfx1250) — compile-verified
//
#include <hip/hip_runtime.h>
#include <hip/hip_bf16.h>

// ---- problem constants (fixed by the reference) ----
static constexpr int  N_ = 8, S_ = 6, C_ = 4, H_ = 512, W_ = 512;
static constexpr int  HW  = H_ * W_;                 // 262144 = 2^18
static constexpr long CHW = (long)C_ * HW;           // 1048576
static constexpr int  P4  = N_ * S_ * HW / 4;        // 3,145,728 quads
static constexpr int  BLOCKS  = 2048;
static constexpr int  THREADS = 256;                 // 8 wave32 per block
static constexpr int  NTHREADS = BLOCKS * THREADS;   // 524,288
static constexpr int  ITERS   = P4 / NTHREADS;       // exactly 6 (no remainder)
static_assert(P4 % NTHREADS == 0, "uniform trip count required");

static constexpr float LOG2E = 1.4426950408889634f;  // log2(e)
static constexpr float LN2   = 0.6931471805599453f;  // ln(2)

typedef __attribute__((ext_vector_type(2))) float v2f;
typedef __attribute__((ext_vector_type(8))) float v8f;
typedef __attribute__((ext_vector_type(4))) float fx4;
typedef __attribute__((ext_vector_type(4))) int   ix4;

// Branchless bitwise select: m must be all-ones or all-zeros. Lowers to v_bfi/bitop3.
__device__ __forceinline__ float bitsel(int m, float x, float y) {
  return __builtin_bit_cast(
      float, (__builtin_bit_cast(int, x) & m) | (__builtin_bit_cast(int, y) & ~m));
}

// Wave32 sum via V_WMMA_F32_16X16X4_F32 with B == ones.
// A layout (32-bit A, 16x4): VGPR0 lanes0-15 -> K=0, lanes16-31 -> K=2; M = lane%16.
// With a = {p, 0}: D[m][n] = p_m + p_{m+16} for every n.
// C/D layout: lane L<16 holds rows 0..7 in c[0..7]; lane L>=16 holds rows 8..15.
// sum(c[0..7]) is a half-wave row-group sum; one ds_swizzle xor-16 swap
// (offset 0x401F: xor=0x10, and=0x1F) completes the full 32-lane sum.
__device__ __forceinline__ float wave_reduce_wmma(float p) {
  v2f a; a[0] = p;    a[1] = 0.0f;
  v2f b; b[0] = 1.0f; b[1] = 1.0f;
  v8f c = {};
  c = __builtin_amdgcn_wmma_f32_16x16x4_f32(false, a, false, b, (short)0, c,
                                            false, false);
  float s = ((c[0] + c[1]) + (c[2] + c[3])) + ((c[4] + c[5]) + (c[6] + c[7]));
  const int swapped =
      __builtin_amdgcn_ds_swizzle(__builtin_bit_cast(int, s), 0x401F);
  return s + __builtin_bit_cast(float, swapped);  // uniform across the wave
}

__global__ void __launch_bounds__(THREADS)
ce_partial_kernel(const float* __restrict__ preds,
                  const int*   __restrict__ tgt,
                  float*       __restrict__ part) {
  __shared__ float wpart[THREADS / 32];

  const int tid = blockIdx.x * THREADS + threadIdx.x;

  float acc = 0.0f;
#pragma unroll
  for (int k = 0; k < ITERS; ++k) {
    const int q   = tid + k * NTHREADS;        // always < P4: uniform, no check
    const int ns  = q >> 16;                   // q / (HW/4), HW/4 = 65536
    const int hw4 = (q & 0xFFFF) << 2;         // position within the HW plane
    const float* base = preds + (long)ns * CHW + hw4;

    // 4 positions x 4 channels: four coalesced 16B non-temporal loads
    const fx4 x0 = __builtin_nontemporal_load((const fx4*)(base));
    const fx4 x1 = __builtin_nontemporal_load((const fx4*)(base + HW));
    const fx4 x2 = __builtin_nontemporal_load((const fx4*)(base + 2 * HW));
    const fx4 x3 = __builtin_nontemporal_load((const fx4*)(base + 3 * HW));
    const ix4 t4 = __builtin_nontemporal_load((const ix4*)(tgt + (long)q * 4));

#pragma unroll
    for (int j = 0; j < 4; ++j) {
      const float a = x0[j], b = x1[j], c = x2[j], d = x3[j];
      // base-2 logsumexp, raw HW transcendentals (args are range-safe here)
      const float ya = a * LOG2E, yb = b * LOG2E, yc = c * LOG2E, yd = d * LOG2E;
      const float my = fmaxf(fmaxf(ya, yb), fmaxf(yc, yd));
      const float es = __builtin_amdgcn_exp2f(ya - my) +
                       __builtin_amdgcn_exp2f(yb - my) +
                       __builtin_amdgcn_exp2f(yc - my) +
                       __builtin_amdgcn_exp2f(yd - my);
      const float lse = (my + __builtin_amdgcn_logf(es)) * LN2;
      // branchless 4-way gather of x[target]
      const int t   = t4[j];
      const int m01 = -(t & 1);          // all-ones if t is odd
      const int m23 = -((t >> 1) & 1);   // all-ones if t >= 2
      const float xt = bitsel(m23, bitsel(m01, d, c), bitsel(m01, b, a));
      acc += lse - xt;                   // nll = logsumexp - x[target]
    }
  }

  // wave reduction on the matrix pipe (EXEC all-ones: fully uniform code above)
  const float ws = wave_reduce_wmma(acc);
  if ((threadIdx.x & 31) == 0) wpart[threadIdx.x >> 5] = ws;
  __syncthreads();
  if (threadIdx.x == 0) {
    float bsum = 0.0f;
#pragma unroll
    for (int i = 0; i < THREADS / 32; ++i) bsum += wpart[i];  // fixed order
    part[blockIdx.x] = bsum;
  }
}

__global__ void __launch_bounds__(THREADS)
ce_final_kernel(const float* __restrict__ part, float* __restrict__ out) {
  __shared__ float wpart[THREADS / 32];

  float acc = 0.0f;
#pragma unroll
  for (int k = 0; k < BLOCKS / THREADS; ++k)   // exactly 8, uniform
    acc += part[threadIdx.x + k * THREADS];

  const float ws = wave_reduce_wmma(acc);
  if ((threadIdx.x & 31) == 0) wpart[threadIdx.x >> 5] = ws;
  __syncthreads();
  if (threadIdx.x == 0) {
    float total = 0.0f;
#pragma unroll
    for (int i = 0; i < THREADS / 32; ++i) total += wpart[i];
    out[0] = total * (1.0f / (float)N_);       // sum over s,h,w of mean over n
  }
}

extern "C" void kernel_launch(void* const* d_in, const int* in_sizes, int n_in,
                              void* d_out, int out_size, void* d_ws, size_t ws_size,
                              hipStream_t stream) {
  const float* preds = (const float*)d_in[0];
  const int*   tgt   = (const int*)d_in[1];
  float*       part  = (float*)d_ws;           // BLOCKS floats = 8 KB scratch

  ce_partial_kernel<<<BLOCKS, THREADS, 0, stream>>>(preds, tgt, part);
  ce_final_kernel<<<1, THREADS, 0, stream>>>(part, (float*)d_out);
}